// DeepLinear_23965917511984
// MI455X (gfx1250) — compile-verified
//
#include <hip/hip_runtime.h>
#include <math.h>

#define B_ 64
#define D_ 512
#define U_ 512
#define NSB 128                      // blocks for the two stat passes
#define PPB ((D_*U_)/NSB)            // 2048 (d,u) pairs per stat block

static constexpr float EPS = 1e-5f;

typedef float v2f __attribute__((ext_vector_type(2)));
typedef float v8f __attribute__((ext_vector_type(8)));

__device__ __forceinline__ float lrelu(float t){ return t >= 0.f ? t : 0.01f*t; }

__device__ __forceinline__ float block_reduce256(float v, float* sm){
  int t = threadIdx.x;
  sm[t] = v; __syncthreads();
  #pragma unroll
  for (int s = 128; s > 0; s >>= 1){
    if (t < s) sm[t] += sm[t+s];
    __syncthreads();
  }
  float r = sm[0]; __syncthreads();
  return r;
}

// ---- workspace layout (float offsets) ----
#define WS_XNT   0        // [D][B] transposed xn           (32768)
#define WS_W1S   32768    // per-d  sum w1                  (512)
#define WS_W2S   33280    // per-d  sum w1^2                (512)
#define WS_WBS   33792    // per-d  sum w1*b1               (512)
#define WS_B1D   34304    // per-d  sum b1                  (512)
#define WS_B2D   34816    // per-d  sum b1^2                (512)
#define WS_GU    35328    // per-u  sum_d g3                (512)
#define WS_E3U   35840    // per-u  sum_d be3               (512)
#define WS_SXN   36352    // per-b  sum_d xn                (64)
#define WS_SCAL  36416    // [B1s, B2s]                     (2)
#define WS_M1    36480    // (64)
#define WS_INV1  36544    // (64)
#define WS_P2    36608    // stage2 partials [NSB][64][2]   (16384)
#define WS_M2    52992
#define WS_INV2  53056
#define WS_P3    53120    // stage3 partials                (16384)
#define WS_M3    69504
#define WS_INV3  69568
// total 69632 floats = 278528 bytes

// K0a: per-d moments of w1/b1 over (u,c)
__global__ void k0a(const float* __restrict__ w1, const float* __restrict__ b1, float* ws){
  int d = blockIdx.x, t = threadIdx.x;
  const float* wp = w1 + d*(U_*2);
  const float* bp = b1 + d*(U_*2);
  float sw=0.f, sw2=0.f, swb=0.f, sb=0.f, sb2=0.f;
  for (int i = t; i < U_*2; i += 256){
    float w = wp[i], bb = bp[i];
    sw += w; sw2 += w*w; swb += w*bb; sb += bb; sb2 += bb*bb;
  }
  __shared__ float sm[256];
  float r;
  r = block_reduce256(sw , sm); if (!t) ws[WS_W1S + d] = r;
  r = block_reduce256(sw2, sm); if (!t) ws[WS_W2S + d] = r;
  r = block_reduce256(swb, sm); if (!t) ws[WS_WBS + d] = r;
  r = block_reduce256(sb , sm); if (!t) ws[WS_B1D + d] = r;
  r = block_reduce256(sb2, sm); if (!t) ws[WS_B2D + d] = r;
}

// K0c: per-u sums of g3, be3 over d
__global__ void k0c(const float* __restrict__ g3, const float* __restrict__ be3, float* ws){
  int u = blockIdx.x*256 + threadIdx.x;
  float sg = 0.f, se = 0.f;
  for (int d = 0; d < D_; ++d){ sg += g3[d*U_ + u]; se += be3[d*U_ + u]; }
  ws[WS_GU + u] = sg; ws[WS_E3U + u] = se;
}

// K1: LayerNorm0 -> xnT [d][b], Sxn[b]
__global__ void k1(const float* __restrict__ x, const float* __restrict__ g0,
                   const float* __restrict__ be0, float* ws){
  int b = blockIdx.x, t = threadIdx.x;
  float v0 = x[b*D_ + t], v1 = x[b*D_ + t + 256];
  __shared__ float sm[256];
  float mean = block_reduce256(v0 + v1, sm) * (1.0f/(float)D_);
  float d0 = v0 - mean, d1 = v1 - mean;
  float var = block_reduce256(d0*d0 + d1*d1, sm) * (1.0f/(float)D_);
  float rstd = rsqrtf(var + EPS);
  float xn0 = d0*rstd*g0[t]     + be0[t];
  float xn1 = d1*rstd*g0[t+256] + be0[t+256];
  ws[WS_XNT + t*B_ + b]       = xn0;
  ws[WS_XNT + (t+256)*B_ + b] = xn1;
  float s = block_reduce256(xn0 + xn1, sm);
  if (!t) ws[WS_SXN + b] = s;
}

// K1b: scalar sums of b1, b1^2
__global__ void k1b(float* ws){
  int t = threadIdx.x;
  __shared__ float sm[256];
  float r1 = block_reduce256(ws[WS_B1D + t] + ws[WS_B1D + t + 256], sm);
  float r2 = block_reduce256(ws[WS_B2D + t] + ws[WS_B2D + t + 256], sm);
  if (!t){ ws[WS_SCAL] = r1; ws[WS_SCAL+1] = r2; }
}

// K2: stage-1 LN stats via WMMA f32 16x16x4 (branch-free B fragment).
// P[b,0] = sum_d xn*W1s ; P[b,1] = sum_d xn^2*W2s + 2*xn*WB
__global__ void k2(float* ws){
  const float* xnT = ws + WS_XNT;
  const float* W1s = ws + WS_W1S;
  const float* W2s = ws + WS_W2S;
  const float* WB  = ws + WS_WBS;
  int lane = threadIdx.x & 31;
  int wv   = threadIdx.x >> 5;          // 4 waves, 16 batches each
  int n    = lane & 15;
  bool hi  = lane >= 16;
  int row  = wv*16 + n;                 // batch row of A this lane feeds
  // 0/1 column masks, hoisted: col0 takes W1s on even K rows; col1 takes
  // 2*WB on even K rows and W2s on odd K rows. Other columns contribute 0.
  float mN0 = (n == 0) ? 1.f : 0.f;
  float mN1 = (n == 1) ? 1.f : 0.f;
  v8f acc = {};
  for (int tt = 0; tt < 256; ++tt){     // K chunks of 4 (K_total = 2*D)
    int d = 2*tt + (hi ? 1 : 0);
    float xv  = xnT[d*B_ + row];
    float w1s = W1s[d];
    float w2s = W2s[d];
    float wb2 = 2.f*WB[d];
    v2f a; a.x = xv; a.y = xv*xv;       // A[row, k_even], A[row, k_odd]
    v2f bf;
    bf.x = w1s*mN0 + wb2*mN1;           // B[k_even, n]
    bf.y = w2s*mN1;                     // B[k_odd,  n]
    acc = __builtin_amdgcn_wmma_f32_16x16x4_f32(false, a, false, bf,
                                                (short)0, acc, false, false);
  }
  __shared__ float P0[B_], P1[B_];
  if (n <= 1){
    #pragma unroll
    for (int r = 0; r < 8; ++r){
      int bb = wv*16 + r + (hi ? 8 : 0);
      if (n == 0) P0[bb] = acc[r]; else P1[bb] = acc[r];
    }
  }
  __syncthreads();
  if (threadIdx.x < B_){
    int bb = threadIdx.x;
    double N1 = (double)D_ * U_ * 2.0;
    double m1 = ((double)P0[bb] + (double)ws[WS_SCAL]) / N1;
    double e2 = ((double)P1[bb] + (double)ws[WS_SCAL+1]) / N1;
    double var = e2 - m1*m1;
    ws[WS_M1 + bb]   = (float)m1;
    ws[WS_INV1 + bb] = (float)(1.0 / sqrt(var + (double)EPS));
  }
}

// K3: pass A — stats for LN2 (sum / sumsq of l21,l22 per batch), partials per block
__global__ void k3(const float* __restrict__ w1,  const float* __restrict__ b1,
                   const float* __restrict__ g1,  const float* __restrict__ be1,
                   const float* __restrict__ w21, const float* __restrict__ w22,
                   const float* __restrict__ b21, const float* __restrict__ b22,
                   float* ws){
  int t = threadIdx.x, b = t & 63, j = t >> 6;
  const float* xnT = ws + WS_XNT;
  float m1 = ws[WS_M1 + b], iv1 = ws[WS_INV1 + b];
  float s = 0.f, s2 = 0.f;
  int base = blockIdx.x * PPB;
  for (int i = 0; i < PPB/4; ++i){
    int p = base + i*4 + j;             // p = d*U + u
    int d = p >> 9;
    if (p + 256 < D_*U_){
      __builtin_prefetch(w1  + 2*(p+256), 0, 3);
      __builtin_prefetch(w21 + 2*(p+256), 0, 3);
    }
    float xv = xnT[d*B_ + b];
    float n1a = (xv*w1[2*p]   + b1[2*p]   - m1)*iv1*g1[2*p]   + be1[2*p];
    float n1b = (xv*w1[2*p+1] + b1[2*p+1] - m1)*iv1*g1[2*p+1] + be1[2*p+1];
    float l1a = lrelu(n1a), l1b = lrelu(n1b);
    float l21 = l1a*w21[2*p] + l1b*w21[2*p+1] + b21[p];
    float l22 = l1a*w22[2*p] + l1b*w22[2*p+1] + b22[p];
    s  += l21 + l22;
    s2 += l21*l21 + l22*l22;
  }
  __shared__ float sm[512];
  sm[t] = s; sm[256 + t] = s2; __syncthreads();
  if (j == 0){
    float S = 0.f, S2 = 0.f;
    #pragma unroll
    for (int k = 0; k < 4; ++k){ S += sm[b + 64*k]; S2 += sm[256 + b + 64*k]; }
    ws[WS_P2 + (blockIdx.x*B_ + b)*2 + 0] = S;
    ws[WS_P2 + (blockIdx.x*B_ + b)*2 + 1] = S2;
  }
}

// K4: pass B — stats for LN3 (sum / sumsq of l3 per batch)
__global__ void k4(const float* __restrict__ w1,  const float* __restrict__ b1,
                   const float* __restrict__ g1,  const float* __restrict__ be1,
                   const float* __restrict__ w21, const float* __restrict__ w22,
                   const float* __restrict__ b21, const float* __restrict__ b22,
                   const float* __restrict__ g2,  const float* __restrict__ be2,
                   const float* __restrict__ w3,  const float* __restrict__ b3,
                   float* ws){
  int t = threadIdx.x, b = t & 63, j = t >> 6;
  const float* xnT = ws + WS_XNT;
  float m1 = ws[WS_M1 + b], iv1 = ws[WS_INV1 + b];
  float m2 = ws[WS_M2 + b], iv2 = ws[WS_INV2 + b];
  float s = 0.f, s2 = 0.f;
  int base = blockIdx.x * PPB;
  for (int i = 0; i < PPB/4; ++i){
    int p = base + i*4 + j;
    int d = p >> 9;
    if (p + 256 < D_*U_){
      __builtin_prefetch(w1 + 2*(p+256), 0, 3);
      __builtin_prefetch(w3 + 2*(p+256), 0, 3);
    }
    float xv = xnT[d*B_ + b];
    float n1a = (xv*w1[2*p]   + b1[2*p]   - m1)*iv1*g1[2*p]   + be1[2*p];
    float n1b = (xv*w1[2*p+1] + b1[2*p+1] - m1)*iv1*g1[2*p+1] + be1[2*p+1];
    float l1a = lrelu(n1a), l1b = lrelu(n1b);
    float l21 = l1a*w21[2*p] + l1b*w21[2*p+1] + b21[p];
    float l22 = l1a*w22[2*p] + l1b*w22[2*p+1] + b22[p];
    float l2a = lrelu((l21 - m2)*iv2*g2[2*p]   + be2[2*p]);
    float l2b = lrelu((l22 - m2)*iv2*g2[2*p+1] + be2[2*p+1]);
    float l3  = l2a*w3[2*p] + l2b*w3[2*p+1] + b3[p];
    s += l3; s2 += l3*l3;
  }
  __shared__ float sm[512];
  sm[t] = s; sm[256 + t] = s2; __syncthreads();
  if (j == 0){
    float S = 0.f, S2 = 0.f;
    #pragma unroll
    for (int k = 0; k < 4; ++k){ S += sm[b + 64*k]; S2 += sm[256 + b + 64*k]; }
    ws[WS_P3 + (blockIdx.x*B_ + b)*2 + 0] = S;
    ws[WS_P3 + (blockIdx.x*B_ + b)*2 + 1] = S2;
  }
}

// finalize LN stats from per-block partials (deterministic fixed-order sum)
__global__ void kfin(float* ws, int part_off, int m_off, int inv_off, double N){
  int b = threadIdx.x;  // 64 threads
  double S = 0.0, S2 = 0.0;
  for (int k = 0; k < NSB; ++k){
    S  += (double)ws[part_off + (k*B_ + b)*2 + 0];
    S2 += (double)ws[part_off + (k*B_ + b)*2 + 1];
  }
  double m = S / N;
  double var = S2 / N - m*m;
  ws[m_off + b]   = (float)m;
  ws[inv_off + b] = (float)(1.0 / sqrt(var + (double)EPS));
}

// K5: pass C — A[b,u] = sum_d l3*g3, then final output
__global__ void k5(const float* __restrict__ w1,  const float* __restrict__ b1,
                   const float* __restrict__ g1,  const float* __restrict__ be1,
                   const float* __restrict__ w21, const float* __restrict__ w22,
                   const float* __restrict__ b21, const float* __restrict__ b22,
                   const float* __restrict__ g2,  const float* __restrict__ be2,
                   const float* __restrict__ w3,  const float* __restrict__ b3,
                   const float* __restrict__ g3,  const float* __restrict__ bias,
                   const float* ws, float* __restrict__ out){
  int t = threadIdx.x, b = t & 63, slot = t >> 6;
  int u = blockIdx.x*4 + slot;
  const float* xnT = ws + WS_XNT;
  float m1 = ws[WS_M1 + b], iv1 = ws[WS_INV1 + b];
  float m2 = ws[WS_M2 + b], iv2 = ws[WS_INV2 + b];
  float A = 0.f;
  for (int d = 0; d < D_; ++d){
    int p = d*U_ + u;
    if (d + 8 < D_){
      __builtin_prefetch(w1 + 2*(p + 8*U_), 0, 3);
      __builtin_prefetch(w3 + 2*(p + 8*U_), 0, 3);
    }
    float xv = xnT[d*B_ + b];
    float n1a = (xv*w1[2*p]   + b1[2*p]   - m1)*iv1*g1[2*p]   + be1[2*p];
    float n1b = (xv*w1[2*p+1] + b1[2*p+1] - m1)*iv1*g1[2*p+1] + be1[2*p+1];
    float l1a = lrelu(n1a), l1b = lrelu(n1b);
    float l21 = l1a*w21[2*p] + l1b*w21[2*p+1] + b21[p];
    float l22 = l1a*w22[2*p] + l1b*w22[2*p+1] + b22[p];
    float l2a = lrelu((l21 - m2)*iv2*g2[2*p]   + be2[2*p]);
    float l2b = lrelu((l22 - m2)*iv2*g2[2*p+1] + be2[2*p+1]);
    float l3  = l2a*w3[2*p] + l2b*w3[2*p+1] + b3[p];
    A += l3 * g3[p];
  }
  float m3 = ws[WS_M3 + b], iv3 = ws[WS_INV3 + b];
  float o = iv3*(A - m3*ws[WS_GU + u]) + ws[WS_E3U + u] + ws[WS_SXN + b] + bias[u];
  out[b*U_ + u] = lrelu(o);
}

extern "C" void kernel_launch(void* const* d_in, const int* in_sizes, int n_in,
                              void* d_out, int out_size, void* d_ws, size_t ws_size,
                              hipStream_t stream) {
  const float* x    = (const float*)d_in[0];
  const float* w1   = (const float*)d_in[1];
  const float* b1   = (const float*)d_in[2];
  const float* w21  = (const float*)d_in[3];
  const float* w22  = (const float*)d_in[4];
  const float* b21  = (const float*)d_in[5];
  const float* b22  = (const float*)d_in[6];
  const float* w3   = (const float*)d_in[7];
  const float* b3   = (const float*)d_in[8];
  const float* bias = (const float*)d_in[9];
  const float* g0   = (const float*)d_in[10];
  const float* be0  = (const float*)d_in[11];
  const float* g1   = (const float*)d_in[12];
  const float* be1  = (const float*)d_in[13];
  const float* g2   = (const float*)d_in[14];
  const float* be2  = (const float*)d_in[15];
  const float* g3   = (const float*)d_in[16];
  const float* be3  = (const float*)d_in[17];
  float* ws  = (float*)d_ws;
  float* out = (float*)d_out;

  k0a<<<D_, 256, 0, stream>>>(w1, b1, ws);
  k0c<<<U_/256, 256, 0, stream>>>(g3, be3, ws);
  k1 <<<B_, 256, 0, stream>>>(x, g0, be0, ws);
  k1b<<<1, 256, 0, stream>>>(ws);
  k2 <<<1, 128, 0, stream>>>(ws);
  k3 <<<NSB, 256, 0, stream>>>(w1, b1, g1, be1, w21, w22, b21, b22, ws);
  kfin<<<1, B_, 0, stream>>>(ws, WS_P2, WS_M2, WS_INV2, (double)D_*U_*2.0);
  k4 <<<NSB, 256, 0, stream>>>(w1, b1, g1, be1, w21, w22, b21, b22, g2, be2, w3, b3, ws);
  kfin<<<1, B_, 0, stream>>>(ws, WS_P3, WS_M3, WS_INV3, (double)D_*U_);
  k5 <<<NSB, 256, 0, stream>>>(w1, b1, g1, be1, w21, w22, b21, b22, g2, be2, w3, b3,
                               g3, bias, ws, out);
}